// WaveRNN_51032801411829
// MI455X (gfx1250) — compile-verified
//
#include <hip/hip_runtime.h>

// ---------------------------------------------------------------------------
// WaveRNN: y[b,t,p] = probes of a linear 2D wave sim driven by x[b,t].
// The sim is LTI in the injection => y = x (*) r  (causal convolution), where
// r[p,d] is the impulse response at probe p. So:
//   1) init:      precompute c1, c2, c^2*DT^2; zero state + sync counter
//   2) impulse:   ONE persistent kernel, B=1, 256 steps, atomic grid-sync
//                 (state = 256KB, fully L2 resident; avoids 256 launches)
//   3) conv:      Y_p(16x256) = X(16x256) * Toeplitz(r_p)(256x256) in f32
//                 via V_WMMA_F32_16X16X4_F32, K-loop unrolled x4 with two
//                 independent accumulators so WMMAs issue back-to-back.
// ---------------------------------------------------------------------------

#define NXY        256
#define NCELL      (NXY * NXY)
#define T_STEPS    256
#define BATCH      16
#define NP         3
#define SIM_BLOCKS 16
#define SIM_THREADS 256
#define SIM_TOTAL  (SIM_BLOCKS * SIM_THREADS)           // 4096 threads
#define CELLS_PER_THREAD (NCELL / SIM_TOTAL)            // 16

typedef float v2f __attribute__((ext_vector_type(2)));
typedef float v8f __attribute__((ext_vector_type(8)));

// Workspace layout (floats):
//  [0*NCELL .. 3*NCELL)  : fields f0,f1,f2 (rotating h2/h1/hn)
//  [3*NCELL .. 4*NCELL)  : c1   = 1/(1+0.25*b)
//  [4*NCELL .. 5*NCELL)  : c2   = 1-0.25*b
//  [5*NCELL .. 6*NCELL)  : csq  = 0.25*c*c
//  [6*NCELL .. +NP*T)    : r[p][t] impulse response
//  then one int          : grid-sync counter
#define WS_R_OFF   (6 * NCELL)
#define WS_CNT_OFF (6 * NCELL + NP * T_STEPS)

__global__ void init_kernel(const float* __restrict__ c,
                            const float* __restrict__ b,
                            float* __restrict__ ws) {
    int idx = blockIdx.x * blockDim.x + threadIdx.x;
    if (idx < NCELL) {
        float bv = b[idx];
        float cv = c[idx];
        ws[3 * NCELL + idx] = 1.0f / (1.0f + 0.25f * bv); // 0.5*DT = 0.25
        ws[4 * NCELL + idx] = 1.0f - 0.25f * bv;
        ws[5 * NCELL + idx] = 0.25f * cv * cv;            // DT^2 = 0.25
        ws[0 * NCELL + idx] = 0.0f;                       // h2 = 0
        ws[1 * NCELL + idx] = 0.0f;                       // h1 = 0
    }
    if (idx < NP * T_STEPS) ws[WS_R_OFF + idx] = 0.0f;
    if (idx == 0) ((int*)(ws + WS_CNT_OFF))[0] = 0;
}

// Persistent B=1 impulse simulation. 256 sequential steps inside one launch;
// inter-step sync via monotonic atomic counter (SIM_BLOCKS co-resident WGs).
__global__ void impulse_sim(const int* __restrict__ px,
                            const int* __restrict__ py,
                            const int* __restrict__ sx,
                            const int* __restrict__ sy,
                            float* __restrict__ ws) {
    float* f[3] = { ws, ws + NCELL, ws + 2 * NCELL };
    const float* c1 = ws + 3 * NCELL;
    const float* c2 = ws + 4 * NCELL;
    const float* cq = ws + 5 * NCELL;
    float* r  = ws + WS_R_OFF;
    int*  cnt = (int*)(ws + WS_CNT_OFF);

    const int tid = blockIdx.x * blockDim.x + threadIdx.x;
    const int src = sx[0] * NXY + sy[0];
    const int pr0 = px[0] * NXY + py[0];
    const int pr1 = px[1] * NXY + py[1];
    const int pr2 = px[2] * NXY + py[2];

    for (int t = 0; t < T_STEPS; ++t) {
        const float* h2 = f[t % 3];
        const float* h1 = f[(t + 1) % 3];
        float*       hn = f[(t + 2) % 3];

        #pragma unroll
        for (int i = 0; i < CELLS_PER_THREAD; ++i) {
            int idx = tid + i * SIM_TOTAL;
            int row = idx >> 8;
            int col = idx & (NXY - 1);
            float hc = h1[idx];
            float up = (row > 0)       ? h1[idx - NXY] : 0.0f;  // zero pad
            float dn = (row < NXY - 1) ? h1[idx + NXY] : 0.0f;
            float lf = (col > 0)       ? h1[idx - 1]   : 0.0f;
            float rt = (col < NXY - 1) ? h1[idx + 1]   : 0.0f;
            float lap = up + dn + lf + rt - 4.0f * hc;
            float v = c1[idx] * (2.0f * hc - c2[idx] * h2[idx] + cq[idx] * lap);
            if (t == 0 && idx == src) v += 1.0f;        // unit impulse
            hn[idx] = v;
            if (idx == pr0) r[0 * T_STEPS + t] = v;     // includes injection
            if (idx == pr1) r[1 * T_STEPS + t] = v;
            if (idx == pr2) r[2 * T_STEPS + t] = v;
        }

        // ---- grid-wide sync (release -> arrive -> acquire) ----
        __threadfence();
        __syncthreads();
        if (threadIdx.x == 0) {
            atomicAdd(cnt, 1);
            int target = SIM_BLOCKS * (t + 1);
            while (atomicAdd(cnt, 0) < target) __builtin_amdgcn_s_sleep(1);
        }
        __syncthreads();
        __threadfence();   // invalidate stale L0/L1 before reading hn next step
    }
}

// Build A (16x4 slice of X) and B (4x16 Toeplitz slice of r_p) operands for
// one K-chunk starting at kb, per the ISA 7.12.2 f32 layouts:
//   A: lanes 0-15 hold M=0..15 with VGPR{0,1}=K{kb,kb+1}; lanes 16-31 K{kb+2,kb+3}
//   B: VGPR0 = rows K0(l0-15)/K2(l16-31), VGPR1 = rows K1/K3, column N = lane&15
__device__ __forceinline__ void make_ab(const float* xs, const float* rp,
                                        int m, int hi, int T0, int kb,
                                        v2f& a, v2f& bv) {
    a.x = xs[m * T_STEPS + kb + 2 * hi];
    a.y = xs[m * T_STEPS + kb + 2 * hi + 1];
    int d0 = T0 + m - (kb + 2 * hi);     // B[2*hi    ][n] = r_p[t - tau]
    int d1 = d0 - 1;                     // B[2*hi + 1][n]
    bv.x = (d0 >= 0) ? rp[d0] : 0.0f;    // zero above the diagonal (tau > t)
    bv.y = (d1 >= 0) ? rp[d1] : 0.0f;
}

// Causal convolution as 3 f32 GEMMs with lower-triangular Toeplitz operands.
// One wave per (probe p, 16-wide t tile): D(16x16) += A(16x4) x B(4x16).
// K-step count is 4*(ttile+1) -> exact unroll by 4; acc0/acc1 alternate so
// consecutive v_wmma ops have no D->C serial dependency.
__global__ void conv_wmma(const float* __restrict__ x,
                          float* __restrict__ out,
                          const float* __restrict__ rr) {
    __shared__ float xs[BATCH * T_STEPS];  // 16 KB
    __shared__ float rs[NP * T_STEPS];     //  3 KB
    for (int i = threadIdx.x; i < BATCH * T_STEPS; i += blockDim.x) xs[i] = x[i];
    for (int i = threadIdx.x; i < NP * T_STEPS;    i += blockDim.x) rs[i] = rr[i];
    __syncthreads();

    const int wave = blockIdx.x * (blockDim.x / 32) + (threadIdx.x >> 5); // 0..47
    const int lane = threadIdx.x & 31;
    const int p     = wave / 16;          // probe 0..2
    const int ttile = wave % 16;          // tile of 16 output timesteps
    const int T0    = ttile * 16;
    const int m     = lane & 15;          // A row (batch) / B,D column
    const int hi    = lane >> 4;          // 0: K{0,1}, 1: K{2,3}

    const float* rp = rs + p * T_STEPS;
    v8f acc0 = {};                        // two independent accumulator chains
    v8f acc1 = {};

    for (int kb0 = 0; kb0 < T0 + 16; kb0 += 16) {   // (ttile+1) iterations
        v2f a0, b0, a1, b1, a2, b2, a3, b3;
        make_ab(xs, rp, m, hi, T0, kb0 +  0, a0, b0);
        make_ab(xs, rp, m, hi, T0, kb0 +  4, a1, b1);
        make_ab(xs, rp, m, hi, T0, kb0 +  8, a2, b2);
        make_ab(xs, rp, m, hi, T0, kb0 + 12, a3, b3);
        acc0 = __builtin_amdgcn_wmma_f32_16x16x4_f32(false, a0, false, b0,
                                                     (short)0, acc0, false, false);
        acc1 = __builtin_amdgcn_wmma_f32_16x16x4_f32(false, a1, false, b1,
                                                     (short)0, acc1, false, false);
        acc0 = __builtin_amdgcn_wmma_f32_16x16x4_f32(false, a2, false, b2,
                                                     (short)0, acc0, false, false);
        acc1 = __builtin_amdgcn_wmma_f32_16x16x4_f32(false, a3, false, b3,
                                                     (short)0, acc1, false, false);
    }

    // Scatter D: out[b][t][p], b = vgpr + 8*hi, t = T0 + (lane&15)
    #pragma unroll
    for (int v = 0; v < 8; ++v) {
        int b = v + 8 * hi;
        int t = T0 + m;
        out[(b * T_STEPS + t) * NP + p] = acc0[v] + acc1[v];
    }
}

extern "C" void kernel_launch(void* const* d_in, const int* in_sizes, int n_in,
                              void* d_out, int out_size, void* d_ws, size_t ws_size,
                              hipStream_t stream) {
    const float* x  = (const float*)d_in[0];   // (16, 256)
    const float* c  = (const float*)d_in[1];   // (256, 256)
    const float* b  = (const float*)d_in[2];   // (256, 256)
    const int*   px = (const int*)d_in[3];     // (3,)
    const int*   py = (const int*)d_in[4];     // (3,)
    const int*   sx = (const int*)d_in[5];     // scalar
    const int*   sy = (const int*)d_in[6];     // scalar
    float* out = (float*)d_out;                // (16, 256, 3)
    float* ws  = (float*)d_ws;

    init_kernel<<<NCELL / 256, 256, 0, stream>>>(c, b, ws);
    impulse_sim<<<SIM_BLOCKS, SIM_THREADS, 0, stream>>>(px, py, sx, sy, ws);
    conv_wmma<<<(NP * 16) / 8, 256, 0, stream>>>(x, out, ws + WS_R_OFF);
}